// SimpleWitCustomOp_82978768158971
// MI455X (gfx1250) — compile-verified
//
#include <hip/hip_runtime.h>

typedef __attribute__((ext_vector_type(2))) float v2f;
typedef __attribute__((ext_vector_type(8))) float v8f;

#define NROWS 8192
#define DIM   2048
#define SCALE 0.1f

// ---------------- Phase 1: column sums of W via WMMA ----------------
// wcol[j] = sum_o W[o, j].  D = ones(16x4) @ B(4x16) accumulated over K.
// 4-way row split: wave (split, colgroup) sums rows [split*512, split*512+512)
// for columns [colgroup*16, colgroup*16+16). Partials -> wp[4][2048].
__global__ void colsum_wmma_kernel(const float* __restrict__ W,
                                   float* __restrict__ wp) {
    int gtid = blockIdx.x * blockDim.x + threadIdx.x;
    int wave = gtid >> 5;
    int lane = threadIdx.x & 31;
    int half = lane >> 4;        // 0: K rows {k0,k0+1}, 1: {k0+2,k0+3}
    int n    = lane & 15;        // column within 16-wide tile

    int split    = wave >> 7;    // 0..3
    int colgroup = wave & 127;   // 0..127
    int j = colgroup * 16 + n;

    v2f a; a.x = 1.0f; a.y = 1.0f;   // A = all-ones -> D[m][n] = sum_k B[k][n]
    v8f c = {};

    int kbase = split * 512;
    for (int k0 = kbase; k0 < kbase + 512; k0 += 4) {
        int r = k0 + 2 * half;
        v2f b;
        b.x = W[(long)r * DIM + j];         // B[k=2h  ][n]
        b.y = W[(long)(r + 1) * DIM + j];   // B[k=2h+1][n]
        c = __builtin_amdgcn_wmma_f32_16x16x4_f32(
                false, a, false, b, (short)0, c, false, false);
    }
    // D[0][n] lives in VGPR0, lanes 0..15
    if (lane < 16)
        wp[split * DIM + colgroup * 16 + n] = c[0];
}

// ---------------- Phase 2: combine wcol partials + reduce bias ----------------
__global__ void combine_kernel(const float* __restrict__ wp,
                               const float* __restrict__ bias,
                               float* __restrict__ wc,
                               float* __restrict__ sb) {
    int j = blockIdx.x * 256 + threadIdx.x;
    if (j < DIM)
        wc[j] = wp[j] + wp[DIM + j] + wp[2 * DIM + j] + wp[3 * DIM + j];

    if (blockIdx.x == 0) {
        __shared__ float s[256];
        float acc = 0.0f;
        for (int t = threadIdx.x; t < DIM; t += 256) acc += bias[t];
        s[threadIdx.x] = acc;
        __syncthreads();
        for (int off = 128; off > 0; off >>= 1) {
            if (threadIdx.x < off) s[threadIdx.x] += s[threadIdx.x + off];
            __syncthreads();
        }
        if (threadIdx.x == 0) sb[0] = s[0];
    }
}

// ---------------- Phase 3: GEMV out_partial = x @ wcol via WMMA ----------------
// B[k][n] = x[row0+n, k0+k]  (one 8B load per lane per step)
// A[m][k] = wcol[k0+k] broadcast to all 16 rows -> every D row = dot product.
// 4-way K split: wave (rowgroup, split) handles rows [rowgroup*16, +16),
// K range [split*512, +512). Partials -> op[4][8192].
__global__ void gemv_wmma_kernel(const float* __restrict__ x,
                                 const float* __restrict__ wc,
                                 float* __restrict__ op) {
    int gtid = blockIdx.x * blockDim.x + threadIdx.x;
    int wave = gtid >> 5;
    int lane = threadIdx.x & 31;
    int half = lane >> 4;
    int n    = lane & 15;

    int rowgroup = wave >> 2;    // 0..511
    int split    = wave & 3;     // 0..3
    long row = (long)rowgroup * 16 + n;
    const float* xrow = x + row * DIM;

    v8f c = {};
    int kbase = split * 512;
    for (int k0 = kbase; k0 < kbase + 512; k0 += 4) {
        v2f b = *(const v2f*)(xrow + k0 + 2 * half);   // x[row_n, k0+2h .. +1]
        v2f a;
        a.x = wc[k0 + 2 * half];
        a.y = wc[k0 + 2 * half + 1];
        c = __builtin_amdgcn_wmma_f32_16x16x4_f32(
                false, a, false, b, (short)0, c, false, false);
    }
    if (lane < 16)
        op[(long)split * NROWS + rowgroup * 16 + n] = c[0];
}

// ---------------- Phase 4: finalize ----------------
__global__ void finalize_kernel(const float* __restrict__ op,
                                const float* __restrict__ sb,
                                float* __restrict__ out) {
    int r = blockIdx.x * 256 + threadIdx.x;
    if (r < NROWS) {
        float v = op[r] + op[NROWS + r] + op[2 * NROWS + r] + op[3 * NROWS + r];
        out[r] = SCALE * (v + sb[0]);
    }
}

extern "C" void kernel_launch(void* const* d_in, const int* in_sizes, int n_in,
                              void* d_out, int out_size, void* d_ws, size_t ws_size,
                              hipStream_t stream) {
    const float* x = (const float*)d_in[0];   // [8192, 2048]
    const float* W = (const float*)d_in[1];   // [2048, 2048]
    const float* b = (const float*)d_in[2];   // [2048]
    float* out = (float*)d_out;               // [8192]

    float* ws = (float*)d_ws;
    float* wp = ws;                   // [4][2048] wcol partials
    float* wc = ws + 4 * DIM;         // [2048]    combined wcol
    float* sb = ws + 5 * DIM;         // [1]       sum of bias
    float* op = ws + 5 * DIM + 256;   // [4][8192] out partials

    // Phase 1: 512 waves (4 splits x 128 column groups) = 64 blocks x 256
    colsum_wmma_kernel<<<64, 256, 0, stream>>>(W, wp);
    // Phase 2: 2048 threads
    combine_kernel<<<8, 256, 0, stream>>>(wp, b, wc, sb);
    // Phase 3: 2048 waves (512 row groups x 4 K-splits) = 256 blocks x 256
    gemv_wmma_kernel<<<256, 256, 0, stream>>>(x, wc, op);
    // Phase 4: 8192 threads
    finalize_kernel<<<32, 256, 0, stream>>>(op, sb, out);
}